// Linear_67516885893740
// MI455X (gfx1250) — compile-verified
//
#include <hip/hip_runtime.h>
#include <hip/hip_bf16.h>

typedef __attribute__((ext_vector_type(2))) float v2f;
typedef __attribute__((ext_vector_type(8))) float v8f;

#define DIM   144          // 16*(1) + 16*(3) + 16*(5)
#define NTILE 9            // 144 / 16 output-channel tiles
#define ROWS_PER_BLOCK 128 // 8 waves * 16 batch rows

// ---------------------------------------------------------------------------
// Kernel 1: W[i,j] = (sum_k mixing[k,i,j] * weight[k]) * norm[i,j]
// 144*144 = 20736 threads; reads of mixing are coalesced across (i,j) for a
// fixed k; weight[k] is wave-uniform (scalar load). ~64MB traffic, ~3us.
// ---------------------------------------------------------------------------
__global__ __launch_bounds__(256)
void build_W_kernel(const float* __restrict__ mixing,
                    const float* __restrict__ weight,
                    const float* __restrict__ norm,
                    float* __restrict__ W, int Kpaths) {
    const int ij = blockIdx.x * blockDim.x + threadIdx.x;
    if (ij >= DIM * DIM) return;
    float acc = 0.0f;
    for (int k = 0; k < Kpaths; ++k) {
        acc += mixing[(size_t)k * (DIM * DIM) + ij] * weight[k];
    }
    W[ij] = acc * norm[ij];
}

// ---------------------------------------------------------------------------
// Kernel 2: out[z,i] = sum_j A[z,j] * W[i,j]  via V_WMMA_F32_16X16X4_F32.
//
// Block = 256 threads = 8 waves. W (82.9KB) staged in LDS once per block.
// Wave w handles batch rows [z0, z0+16); per wave: 9 f32 16x16 accumulators
// (N = all 144 output channels), K loop of 36 steps of 4 -> 324 WMMAs/wave.
//
// Fragment layouts (ISA 7.12.2, 32-bit):
//   A 16x4 : lane<16 -> M=lane, {K=k, k+1}; lane>=16 -> M=lane-16, {K=k+2, k+3}
//   B 4x16 : lane<16 -> N=lane, {K=k, k+1}; lane>=16 -> N=lane-16, {K=k+2, k+3}
//            B[K,N] = W[N,K] -> 8B contiguous LDS load (W row-major over K)
//   C/D    : VGPR r -> M = r + 8*(lane/16), N = lane%16
// ---------------------------------------------------------------------------
__global__ __launch_bounds__(256)
void gemm_wmma_kernel(const float* __restrict__ A,
                      const float* __restrict__ W,
                      float* __restrict__ out) {
    __shared__ float Wl[DIM * DIM];

    // Cooperative, coalesced stage of W into LDS (81 floats / thread).
    for (int idx = threadIdx.x; idx < DIM * DIM; idx += 256)
        Wl[idx] = W[idx];
    __syncthreads();

    const int wave = threadIdx.x >> 5;
    const int lane = threadIdx.x & 31;
    const int half = lane >> 4;   // 0: K pair {k,k+1}, 1: {k+2,k+3}
    const int l16  = lane & 15;   // M (for A) / N (for B, D)

    const long z0   = (long)blockIdx.x * ROWS_PER_BLOCK + (long)wave * 16;
    const float* __restrict__ arow = A + (z0 + l16) * DIM;

    v8f zero = {};
    v8f acc[NTILE];
#pragma unroll
    for (int t = 0; t < NTILE; ++t) acc[t] = zero;

#pragma unroll 2
    for (int k = 0; k < DIM; k += 4) {
        const int ko = k + 2 * half;                 // even -> 8B aligned
        v2f a = *(const v2f*)(arow + ko);            // A[M=l16, ko..ko+1]
#pragma unroll
        for (int t = 0; t < NTILE; ++t) {
            // B[K=ko..ko+1, N=t*16+l16] = W[t*16+l16, ko..ko+1]
            v2f b = *(const v2f*)(&Wl[(t * 16 + l16) * DIM + ko]);
            acc[t] = __builtin_amdgcn_wmma_f32_16x16x4_f32(
                /*neg_a=*/false, a, /*neg_b=*/false, b,
                /*c_mod=*/(short)0, acc[t],
                /*reuse_a=*/false, /*reuse_b=*/false);
        }
    }

    // D layout: VGPR r holds (M = r + 8*half, N = l16) -> coalesced 64B stores.
#pragma unroll
    for (int t = 0; t < NTILE; ++t) {
        const int n = t * 16 + l16;
#pragma unroll
        for (int r = 0; r < 8; ++r) {
            out[(z0 + r + 8 * half) * DIM + n] = acc[t][r];
        }
    }
}

extern "C" void kernel_launch(void* const* d_in, const int* in_sizes, int n_in,
                              void* d_out, int out_size, void* d_ws, size_t ws_size,
                              hipStream_t stream) {
    const float* features = (const float*)d_in[0];  // [262144, 144]
    const float* weight   = (const float*)d_in[1];  // [768]
    const float* mixing   = (const float*)d_in[2];  // [768, 144, 144]
    const float* norm     = (const float*)d_in[3];  // [144, 144]
    float*       output   = (float*)d_out;          // [262144, 144]
    float*       W        = (float*)d_ws;           // 144*144 floats scratch

    const int Kpaths = in_sizes[1];                 // 768
    const int batch  = in_sizes[0] / DIM;           // 262144

    build_W_kernel<<<(DIM * DIM + 255) / 256, 256, 0, stream>>>(
        mixing, weight, norm, W, Kpaths);

    gemm_wmma_kernel<<<batch / ROWS_PER_BLOCK, 256, 0, stream>>>(
        features, W, output);
}